// GAT_1683627180256
// MI455X (gfx1250) — compile-verified
//
#include <hip/hip_runtime.h>
#include <hip/hip_bf16.h>

// ---------------------------------------------------------------------------
// GAT 2-layer forward for MI455X (gfx1250, wave32).
//  - Dense GEMMs: V_WMMA_F32_16X16X4_F32, 4 accumulators/wave (16x64 strip),
//    A/B staged in LDS via async global->LDS copies (ASYNCcnt double buffer).
//  - Edge softmax: monotonic-uint atomic max + f32 atomic adds (HBM bound).
// ---------------------------------------------------------------------------

typedef __attribute__((ext_vector_type(2))) float v2f;
typedef __attribute__((ext_vector_type(8))) float v8f;

#define IN_F 128
#define HEADS 4
#define HID 64
#define HD (HEADS * HID)       // 256, also HEADS*OUT
#define NEG_SLOPE 0.2f
#define ENC_NEG_INF 0x007FFFFFu

// GEMM tiling
#define KC 16                  // K-chunk staged per double-buffer step
#define BN 256                 // fixed output width (HD)
#define ASTRIDE 18             // padded LDS stride for the A tile (8B aligned)

// ---- float <-> order-preserving uint (for atomicMax on floats) ------------
__device__ __forceinline__ unsigned f2mono(float f) {
  unsigned u = __float_as_uint(f);
  return (u & 0x80000000u) ? ~u : (u | 0x80000000u);
}
__device__ __forceinline__ float mono2f(unsigned u) {
  unsigned v = (u & 0x80000000u) ? (u & 0x7FFFFFFFu) : ~u;
  return __uint_as_float(v);
}

// ---- generic fills --------------------------------------------------------
__global__ void fill_f32_kernel(float* __restrict__ p, float v, int n) {
  int i = blockIdx.x * blockDim.x + threadIdx.x;
  if (i < n) p[i] = v;
}
__global__ void fill_u32_kernel(unsigned* __restrict__ p, unsigned v, int n) {
  int i = blockIdx.x * blockDim.x + threadIdx.x;
  if (i < n) p[i] = v;
}

// ---- async global->LDS helpers (CDNA5, tracked by ASYNCcnt) ---------------
__device__ __forceinline__ void async_g2l_b128(unsigned lds_off, const float* gptr) {
  asm volatile("global_load_async_to_lds_b128 %0, %1, off"
               :: "v"(lds_off), "v"(gptr) : "memory");
}
__device__ __forceinline__ void async_g2l_b64(unsigned lds_off, const float* gptr) {
  asm volatile("global_load_async_to_lds_b64 %0, %1, off"
               :: "v"(lds_off), "v"(gptr) : "memory");
}
__device__ __forceinline__ unsigned lds_offset(const void* p) {
  // generic LDS addresses carry the LDS byte offset in the low 32 bits
  return (unsigned)(uintptr_t)p;
}

// ===========================================================================
// fp32 WMMA GEMM with LDS double buffering.
// Block = 128 threads (4 waves). Block owns one 16-row M-tile and all 256
// output columns. Wave w computes columns [w*64, w*64+64) = 4 N-tiles with
// 4 independent accumulators (breaks the WMMA RAW chain, 4x A-frag reuse).
// Per K-chunk (KC=16): B chunk 16x256 (16KB) + A tile 16x16 staged in LDS
// with async b128/b64 copies; 9 async ops per wave per chunk.
// Requires: M % 16 == 0, K % KC == 0, N == BN.
// ===========================================================================
__global__ __launch_bounds__(128) void wmma_gemm_lds_kernel(
    const float* __restrict__ A, const float* __restrict__ B,
    float* __restrict__ C, int M, int K) {
  __shared__ float lds_b[2][KC * BN];        // 2 x 16KB
  __shared__ float lds_a[2][16 * ASTRIDE];   // 2 x 1.1KB

  const int tid  = threadIdx.x;
  const int wave = tid >> 5;
  const int lane = tid & 31;
  const int half = lane >> 4;   // 0: K pair {0,1}, 1: K pair {2,3}
  const int l    = lane & 15;
  const int m0   = blockIdx.x * 16;
  const int col0 = wave * 64 + l;            // this wave's 4 cols: +0,16,32,48

  // per-chunk async issue: 8 x b128 (B) + 1 x b64 (A) per thread
  auto issue_chunk = [&](int kc, int buf) {
    const float* gB = B + (size_t)kc * BN;   // chunk spans whole rows: contiguous
    unsigned ldsB = lds_offset(&lds_b[buf][0]);
#pragma unroll
    for (int j = 0; j < 8; ++j) {
      int v4 = tid + j * 128;                // 1024 b128 per chunk
      async_g2l_b128(ldsB + v4 * 16, gB + v4 * 4);
    }
    {
      int row = tid >> 3;                    // 16 rows x 8 b64 per row
      int cp  = tid & 7;
      unsigned ldsA = lds_offset(&lds_a[buf][0]) + (row * ASTRIDE + cp * 2) * 4;
      const float* gA = A + (size_t)(m0 + row) * K + kc + cp * 2;
      async_g2l_b64(ldsA, gA);
    }
  };

  v8f acc0 = {}, acc1 = {}, acc2 = {}, acc3 = {};
  const int nchunks = K / KC;

  issue_chunk(0, 0);
  for (int c = 0; c < nchunks; ++c) {
    if (c + 1 < nchunks) {
      issue_chunk((c + 1) * KC, (c + 1) & 1);
      asm volatile("s_wait_asynccnt 9" ::: "memory");  // chunk c complete
    } else {
      asm volatile("s_wait_asynccnt 0" ::: "memory");
    }
    __syncthreads();                                   // all waves' copies visible

    const float* Bs = &lds_b[c & 1][0];
    const float* As = &lds_a[c & 1][0];
#pragma unroll
    for (int kk = 0; kk < KC; kk += 4) {
      v2f a;
      a.x = As[l * ASTRIDE + kk + half * 2];
      a.y = As[l * ASTRIDE + kk + half * 2 + 1];
      const float* bp = Bs + (kk + half * 2) * BN;
      v2f b0, b1, b2, b3;
      b0.x = bp[col0];      b0.y = bp[BN + col0];
      b1.x = bp[col0 + 16]; b1.y = bp[BN + col0 + 16];
      b2.x = bp[col0 + 32]; b2.y = bp[BN + col0 + 32];
      b3.x = bp[col0 + 48]; b3.y = bp[BN + col0 + 48];
      acc0 = __builtin_amdgcn_wmma_f32_16x16x4_f32(false, a, false, b0, (short)0, acc0, false, false);
      acc1 = __builtin_amdgcn_wmma_f32_16x16x4_f32(false, a, false, b1, (short)0, acc1, false, false);
      acc2 = __builtin_amdgcn_wmma_f32_16x16x4_f32(false, a, false, b2, (short)0, acc2, false, false);
      acc3 = __builtin_amdgcn_wmma_f32_16x16x4_f32(false, a, false, b3, (short)0, acc3, false, false);
    }
    __syncthreads();                                   // done reading this buffer
  }

  const int crow0 = m0 + half * 8;
#pragma unroll
  for (int r = 0; r < 8; ++r) {
    float* cp = C + (size_t)(crow0 + r) * BN;
    cp[col0]      = acc0[r];
    cp[col0 + 16] = acc1[r];
    cp[col0 + 32] = acc2[r];
    cp[col0 + 48] = acc3[r];
  }
}

// ---- fallback naive WMMA GEMM (generic shapes) ----------------------------
__global__ void wmma_gemm_f32_kernel(const float* __restrict__ A,
                                     const float* __restrict__ B,
                                     float* __restrict__ C,
                                     int M, int K, int N) {
  const int wave = (blockIdx.x * blockDim.x + threadIdx.x) >> 5;
  const int lane = threadIdx.x & 31;
  const int ntiles = N >> 4;
  const int mt = wave / ntiles;
  const int nt = wave % ntiles;
  if (mt * 16 >= M) return;
  const int half = lane >> 4;
  const int l    = lane & 15;
  int arow = mt * 16 + l;
  if (arow >= M) arow = M - 1;
  const int bcol = nt * 16 + l;
  v8f acc = {};
  for (int k = 0; k < K; k += 4) {
    const float* ap = A + (size_t)arow * K + (k + half * 2);
    v2f a; a.x = ap[0]; a.y = ap[1];
    const float* bp = B + (size_t)(k + half * 2) * N + bcol;
    v2f b; b.x = bp[0]; b.y = bp[N];
    acc = __builtin_amdgcn_wmma_f32_16x16x4_f32(false, a, false, b, (short)0, acc, false, false);
  }
  const int crow0 = mt * 16 + half * 8;
#pragma unroll
  for (int r = 0; r < 8; ++r) {
    int row = crow0 + r;
    if (row < M) C[(size_t)row * N + bcol] = acc[r];
  }
}

// ---- per-(node,head) attention coefficients -------------------------------
__global__ void attn_coef_kernel(const float* __restrict__ z,
                                 const float* __restrict__ al,
                                 const float* __restrict__ ar,
                                 float* __restrict__ el,
                                 float* __restrict__ er, int NH) {
  int idx = blockIdx.x * blockDim.x + threadIdx.x;   // n*HEADS + h
  if (idx >= NH) return;
  int h = idx & (HEADS - 1);
  const float* zp  = z  + (size_t)idx * HID;         // z[n, h, :]
  const float* alp = al + h * HID;
  const float* arp = ar + h * HID;
  float sl = 0.f, sr = 0.f;
#pragma unroll 4
  for (int d = 0; d < HID; ++d) {
    float v = zp[d];
    sl += v * alp[d];
    sr += v * arp[d];
  }
  el[idx] = sl;
  er[idx] = sr;
}

// ---- edge pass 1: e = leaky_relu(el[src]+er[dst]); m[dst] = max -----------
__global__ void edge_logit_max_kernel(const float* __restrict__ el,
                                      const float* __restrict__ er,
                                      const int* __restrict__ src,
                                      const int* __restrict__ dst,
                                      float* __restrict__ e_buf,
                                      unsigned* __restrict__ mmax, int EH) {
  int idx = blockIdx.x * blockDim.x + threadIdx.x;   // eid*HEADS + h
  if (idx >= EH) return;
  int eid = idx >> 2;
  int h   = idx & (HEADS - 1);
  int s = src[eid], d = dst[eid];
  float v = el[s * HEADS + h] + er[d * HEADS + h];
  v = (v > 0.f) ? v : v * NEG_SLOPE;
  e_buf[idx] = v;
  atomicMax(&mmax[d * HEADS + h], f2mono(v));
}

// ---- edge pass 2: ee = exp(e - m[dst]); s[dst] += ee ----------------------
__global__ void edge_exp_sum_kernel(const int* __restrict__ dst,
                                    const unsigned* __restrict__ mmax,
                                    float* __restrict__ e_buf,   // in: e, out: ee
                                    float* __restrict__ ssum, int EH) {
  int idx = blockIdx.x * blockDim.x + threadIdx.x;
  if (idx >= EH) return;
  int eid = idx >> 2;
  int h   = idx & (HEADS - 1);
  int d = dst[eid];
  float m = mono2f(mmax[d * HEADS + h]);
  float v = expf(e_buf[idx] - m);
  e_buf[idx] = v;
  atomicAdd(&ssum[d * HEADS + h], v);
}

// ---- edge pass 3: agg[dst] += z[src] * ee  (one wave32 per edge) ----------
__global__ void edge_scatter_kernel(const float* __restrict__ z,
                                    const int* __restrict__ src,
                                    const int* __restrict__ dst,
                                    const float* __restrict__ ee,
                                    float* __restrict__ agg, int E) {
  int wid  = (blockIdx.x * blockDim.x + threadIdx.x) >> 5;
  int lane = threadIdx.x & 31;
  if (wid >= E) return;
  int s = src[wid], d = dst[wid];
  const float* zp  = z   + (size_t)s * HD;
  float*       ap  = agg + (size_t)d * HD;
  const float* eep = ee  + (size_t)wid * HEADS;
#pragma unroll
  for (int k = 0; k < 8; ++k) {              // 8 * 32 lanes = 256 = HD
    int i = lane + k * 32;
    float a = eep[i >> 6];                   // head = i / HID
    atomicAdd(&ap[i], zp[i] * a);
  }
}

// ---- layer-1 finalize: h1 = ELU(agg / s + b) ------------------------------
__global__ void finalize1_kernel(const float* __restrict__ agg,
                                 const float* __restrict__ ssum,
                                 const float* __restrict__ b,
                                 float* __restrict__ h1, int NHD) {
  int idx = blockIdx.x * blockDim.x + threadIdx.x;   // n*HD + c
  if (idx >= NHD) return;
  int c = idx & (HD - 1);
  int nh = idx >> 6;                                 // n*HEADS + h
  float s = ssum[nh];
  float inv = (s > 0.f) ? 1.f / s : 0.f;
  float v = agg[idx] * inv + b[c];
  h1[idx] = (v > 0.f) ? v : (expf(v) - 1.f);         // ELU(alpha=1)
}

// ---- layer-2 finalize: out[n,o] = mean_h(agg/s + b) -----------------------
__global__ void finalize2_kernel(const float* __restrict__ agg,
                                 const float* __restrict__ ssum,
                                 const float* __restrict__ b,
                                 float* __restrict__ out, int NO) {
  int idx = blockIdx.x * blockDim.x + threadIdx.x;   // n*OUT + o
  if (idx >= NO) return;
  int n = idx >> 6;                                  // OUT = 64
  int o = idx & 63;
  float acc = 0.f;
#pragma unroll
  for (int h = 0; h < HEADS; ++h) {
    float s = ssum[n * HEADS + h];
    float inv = (s > 0.f) ? 1.f / s : 0.f;
    acc += agg[((size_t)n * HEADS + h) * HID + o] * inv + b[h * HID + o];
  }
  out[idx] = acc * (1.f / HEADS);
}

// ---------------------------------------------------------------------------
static inline size_t align256(size_t x) { return (x + 255) & ~(size_t)255; }

static void run_gat_layer(const float* in, int K,
                          const float* W, const float* al, const float* ar,
                          const int* src, const int* dst,
                          float* z, float* agg, float* el, float* er,
                          unsigned* mmax, float* ssum, float* ebuf,
                          int N, int E, hipStream_t stream) {
  const int T = 256;
  fill_f32_kernel<<<(N * HD + T - 1) / T, T, 0, stream>>>(agg, 0.f, N * HD);
  fill_f32_kernel<<<(N * HEADS + T - 1) / T, T, 0, stream>>>(ssum, 0.f, N * HEADS);
  fill_u32_kernel<<<(N * HEADS + T - 1) / T, T, 0, stream>>>(mmax, ENC_NEG_INF, N * HEADS);

  // z = in @ W   (M=N, K, output width HD)
  if ((N % 16 == 0) && (K % KC == 0)) {
    wmma_gemm_lds_kernel<<<N / 16, 128, 0, stream>>>(in, W, z, N, K);
  } else {
    int waves = ((N + 15) / 16) * (HD / 16);
    wmma_gemm_f32_kernel<<<(waves * 32 + T - 1) / T, T, 0, stream>>>(in, W, z, N, K, HD);
  }

  attn_coef_kernel<<<(N * HEADS + T - 1) / T, T, 0, stream>>>(z, al, ar, el, er, N * HEADS);
  edge_logit_max_kernel<<<(E * HEADS + T - 1) / T, T, 0, stream>>>(el, er, src, dst, ebuf, mmax, E * HEADS);
  edge_exp_sum_kernel<<<(E * HEADS + T - 1) / T, T, 0, stream>>>(dst, mmax, ebuf, ssum, E * HEADS);
  edge_scatter_kernel<<<(E * 32 + T - 1) / T, T, 0, stream>>>(z, src, dst, ebuf, agg, E);
}

extern "C" void kernel_launch(void* const* d_in, const int* in_sizes, int n_in,
                              void* d_out, int out_size, void* d_ws, size_t ws_size,
                              hipStream_t stream) {
  const float* x   = (const float*)d_in[0];
  const int*   src = (const int*)  d_in[1];
  const int*   dst = (const int*)  d_in[2];
  const float* W1  = (const float*)d_in[3];
  const float* al1 = (const float*)d_in[4];
  const float* ar1 = (const float*)d_in[5];
  const float* b1  = (const float*)d_in[6];
  const float* W2  = (const float*)d_in[7];
  const float* al2 = (const float*)d_in[8];
  const float* ar2 = (const float*)d_in[9];
  const float* b2  = (const float*)d_in[10];

  const int N = in_sizes[0] / IN_F;   // 50000
  const int E = in_sizes[1];          // 800000

  char* ws = (char*)d_ws;
  size_t off = 0;
  float* z    = (float*)(ws + off); off += align256((size_t)N * HD * 4);
  float* agg  = (float*)(ws + off); off += align256((size_t)N * HD * 4);
  float* h1   = (float*)(ws + off); off += align256((size_t)N * HD * 4);
  float* el   = (float*)(ws + off); off += align256((size_t)N * HEADS * 4);
  float* er   = (float*)(ws + off); off += align256((size_t)N * HEADS * 4);
  unsigned* mmax = (unsigned*)(ws + off); off += align256((size_t)N * HEADS * 4);
  float* ssum = (float*)(ws + off); off += align256((size_t)N * HEADS * 4);
  float* ebuf = (float*)(ws + off); off += align256((size_t)E * HEADS * 4);
  (void)ws_size; (void)n_in; (void)out_size;

  const int T = 256;

  // ---- layer 1: x[N x 128] -> h1[N x 256] ----
  run_gat_layer(x, IN_F, W1, al1, ar1, src, dst,
                z, agg, el, er, mmax, ssum, ebuf, N, E, stream);
  finalize1_kernel<<<(N * HD + T - 1) / T, T, 0, stream>>>(agg, ssum, b1, h1, N * HD);

  // ---- layer 2: h1[N x 256] -> out[N x 64] ----
  run_gat_layer(h1, HD, W2, al2, ar2, src, dst,
                z, agg, el, er, mmax, ssum, ebuf, N, E, stream);
  finalize2_kernel<<<(N * 64 + T - 1) / T, T, 0, stream>>>(agg, ssum, b2, (float*)d_out, N * 64);
}